// MultiSpectralAttentionLayer_18932215841223
// MI455X (gfx1250) — compile-verified
//
#include <hip/hip_runtime.h>
#include <math.h>

typedef __attribute__((ext_vector_type(2))) float v2f;
typedef __attribute__((ext_vector_type(4))) float v4f;
typedef __attribute__((ext_vector_type(8))) float v8f;

#define BB 32
#define CC 256
#define HH 64
#define WW 64
#define KK (HH * WW)          // 4096 (reduction length per (b,c) row)
#define NROWS (BB * CC)       // 8192
#define HID (CC / 4)          // 64
#define FF 4                  // number of DCT frequency planes
#define KSTEPS (KK / 4)       // 1024 WMMA k-steps per full row
#define WAVES_PER_BLOCK 4
#define STEPS_PER_WAVE (KSTEPS / WAVES_PER_BLOCK) // 256

// ---------------------------------------------------------------------------
// DCT basis value: basis[f][h*W+w] = A_H[u_f][h] * A_W[v_f][w]
// A[0][h] = sqrt(1/64) = 0.125 ; A[1][h] = cos(pi*(2h+1)/128) * sqrt(2)/8
// FREQ_SEL = (0,0),(0,1),(1,0),(1,1)
// ---------------------------------------------------------------------------
__device__ __forceinline__ float basis_val(int f, int k) {
    int h = k >> 6;
    int w = k & 63;
    const float s0 = 0.125f;
    const float sc = 0.17677669529663687f; // sqrt(2)/8
    float ch = __cosf(3.14159265358979323846f * (2.0f * (float)h + 1.0f) / 128.0f) * sc;
    float cw = __cosf(3.14159265358979323846f * (2.0f * (float)w + 1.0f) / 128.0f) * sc;
    float r;
    if (f == 0)      r = s0 * s0;
    else if (f == 1) r = s0 * cw;
    else if (f == 2) r = ch * s0;
    else             r = ch * cw;
    return r;
}

// ---------------------------------------------------------------------------
// K0: build B (basis^T) pre-swizzled into the WMMA B-matrix lane layout.
// For wmma_f32_16x16x4_f32, B is 4x16 (KxN): lane L holds rows
// K = 2*(L>>4) + {0,1} for column N = L&15 as a float2. Only N<4 is live.
// Bpacked[step][lane] covers k0 = 4*step.
// ---------------------------------------------------------------------------
__global__ __launch_bounds__(256) void build_bpacked(float* __restrict__ bp) {
    int idx = blockIdx.x * 256 + threadIdx.x;   // 0 .. KSTEPS*32-1
    if (idx >= KSTEPS * 32) return;
    int step = idx >> 5;
    int lane = idx & 31;
    int n    = lane & 15;
    int k    = step * 4 + 2 * (lane >> 4);
    v2f v = {};
    if (n < FF) {
        v.x = basis_val(n, k);
        v.y = basis_val(n, k + 1);
    }
    ((v2f*)bp)[idx] = v;
}

// ---------------------------------------------------------------------------
// K1: spectral[b,c,f] via V_WMMA_F32_16X16X4_F32.
// Block = 128 threads (4 waves). Block owns 16 consecutive (b,c) rows.
// Wave w accumulates k in [w*1024, (w+1)*1024); partials reduced in LDS.
// ---------------------------------------------------------------------------
__global__ __launch_bounds__(128) void spectral_wmma(const float* __restrict__ x,
                                                     const float* __restrict__ bp,
                                                     float* __restrict__ spectral) {
    __shared__ float part[WAVES_PER_BLOCK][16][FF];

    const int tid  = threadIdx.x;
    const int wave = tid >> 5;
    const int lane = tid & 31;
    const int half = lane >> 4;   // which K-pair this lane holds
    const int n    = lane & 15;   // A row (m) and B/C column (n)

    const size_t row  = (size_t)blockIdx.x * 16 + (size_t)n;   // A-matrix row m = lane%16
    const float* arow = x + row * (size_t)KK;
    const int kOff     = wave * (KK / WAVES_PER_BLOCK) + 2 * half;
    const int stepBase = wave * STEPS_PER_WAVE;
    const v2f* bpv = (const v2f*)bp;

    v8f acc = {};
#pragma unroll 4
    for (int t = 0; t < STEPS_PER_WAVE; ++t) {
        v2f a = *(const v2f*)(arow + kOff + 4 * t);      // A: x[row][4t + 2*half + {0,1}]
        v2f b = bpv[(size_t)(stepBase + t) * 32 + lane]; // B: pre-swizzled basis^T
        acc = __builtin_amdgcn_wmma_f32_16x16x4_f32(
            /*neg_a=*/false, a, /*neg_b=*/false, b,
            /*c_mod=*/(short)0, acc, /*reuse_a=*/false, /*reuse_b=*/false);
    }

    // C/D layout: VGPR r holds row M = r + 8*(lane>>4), column N = lane&15.
    if (n < FF) {
#pragma unroll
        for (int r = 0; r < 8; ++r) {
            int m = r + 8 * half;
            part[wave][m][n] = acc[r];
        }
    }
    __syncthreads();

    if (tid < 64) {
        int m = tid >> 2;
        int f = tid & 3;
        float s = part[0][m][f] + part[1][m][f] + part[2][m][f] + part[3][m][f];
        spectral[((size_t)blockIdx.x * 16 + m) * FF + f] = s;
    }
}

// ---------------------------------------------------------------------------
// K2: gate[b,c] = sigmoid( mean_f( w2 @ relu( w1 @ spectral[b] ) ) )
// One block per batch element. Tiny FLOP count; plain VALU is fine.
// ---------------------------------------------------------------------------
__global__ __launch_bounds__(256) void gate_kernel(const float* __restrict__ spectral,
                                                   const float* __restrict__ w1,
                                                   const float* __restrict__ w2,
                                                   float* __restrict__ gate) {
    __shared__ __align__(16) float s[CC][FF];   // 4 KiB
    __shared__ float y1[HID][FF];               // 1 KiB

    const int b = blockIdx.x;
    const int t = threadIdx.x;

    // stage spectral[b] (1024 floats) via float4: 256 threads x 1
    ((v4f*)&s[0][0])[t] = ((const v4f*)(spectral + (size_t)b * CC * FF))[t];
    __syncthreads();

    // fc1 + relu: y1[h][f] = relu( sum_c w1[h][c] * s[c][f] )   (64*4 = 256 outputs)
    {
        int h = t >> 2;
        int f = t & 3;
        float acc = 0.0f;
#pragma unroll 4
        for (int c = 0; c < CC; ++c) acc += w1[h * CC + c] * s[c][f];
        y1[h][f] = fmaxf(acc, 0.0f);
    }
    __syncthreads();

    // fc2 + mean over f + sigmoid: one channel per thread
    {
        int c = t;
        float a0 = 0.f, a1 = 0.f, a2 = 0.f, a3 = 0.f;
#pragma unroll 4
        for (int h = 0; h < HID; ++h) {
            float w = w2[c * HID + h];
            a0 += w * y1[h][0];
            a1 += w * y1[h][1];
            a2 += w * y1[h][2];
            a3 += w * y1[h][3];
        }
        float m = 0.25f * (a0 + a1 + a2 + a3);
        gate[(size_t)b * CC + c] = 1.0f / (1.0f + __expf(-m));
    }
}

// ---------------------------------------------------------------------------
// K3: out[b,c,h,w] = x[b,c,h,w] * gate[b,c]  — pure bandwidth, float4.
// One block per (b,c) row: 256 threads x 4 float4 = 4096 elements.
// ---------------------------------------------------------------------------
__global__ __launch_bounds__(256) void scale_kernel(const float* __restrict__ x,
                                                    const float* __restrict__ gate,
                                                    float* __restrict__ out) {
    const size_t row = blockIdx.x;
    const float g = gate[row];
    const v4f* __restrict__ xin = (const v4f*)(x + row * (size_t)KK);
    v4f* __restrict__ o         = (v4f*)(out + row * (size_t)KK);
    const int t = threadIdx.x;
#pragma unroll
    for (int i = 0; i < 4; ++i) {
        v4f v = xin[t + 256 * i];
        o[t + 256 * i] = v * g;
    }
}

// ---------------------------------------------------------------------------
extern "C" void kernel_launch(void* const* d_in, const int* in_sizes, int n_in,
                              void* d_out, int out_size, void* d_ws, size_t ws_size,
                              hipStream_t stream) {
    const float* x  = (const float*)d_in[0];
    const float* w1 = (const float*)d_in[1];
    const float* w2 = (const float*)d_in[2];
    float* out = (float*)d_out;

    // Workspace layout (all fp32):
    //   [0, 256 KiB)       Bpacked  : KSTEPS*32*2 floats
    //   [256 KiB, 384 KiB) spectral : NROWS*FF floats
    //   [384 KiB, 416 KiB) gate     : NROWS floats
    char* ws = (char*)d_ws;
    float* bpacked  = (float*)(ws);
    float* spectral = (float*)(ws + (size_t)KSTEPS * 32 * 2 * sizeof(float));
    float* gate     = (float*)(ws + (size_t)KSTEPS * 32 * 2 * sizeof(float)
                                  + (size_t)NROWS * FF * sizeof(float));

    // K0: build pre-swizzled B matrix (basis^T in WMMA lane layout)
    build_bpacked<<<(KSTEPS * 32 + 255) / 256, 256, 0, stream>>>(bpacked);

    // K1: spectral projection via f32 WMMA
    spectral_wmma<<<NROWS / 16, 128, 0, stream>>>(x, bpacked, spectral);

    // K2: channel-MLP gate
    gate_kernel<<<BB, 256, 0, stream>>>(spectral, w1, w2, gate);

    // K3: apply gate
    scale_kernel<<<NROWS, 256, 0, stream>>>(x, gate, out);
}